// Net_37331855736934
// MI455X (gfx1250) — compile-verified
//
#include <hip/hip_runtime.h>

// LSTM fused scan for MI455X (gfx1250), wave32.
// One wave = 2 independent 16-batch tiles; 16x v_wmma_f32_16x16x4_f32 per
// tile per timestep (4 gate column-tiles x 4 K-chunks over [h(10)|x(5)|0]).
// Gate nonlinearities use v_exp_f32 + v_rcp_f32 (no IEEE-division expansion).
// Bias splats feed the first WMMA of each chain as C directly (no acc movs).

typedef __attribute__((ext_vector_type(2))) float v2f;
typedef __attribute__((ext_vector_type(8))) float v8f;

#define B_SZ 2048
#define T_SZ 4096
#define I_SZ 5
#define H_SZ 10
#define NT   2   // batch tiles (16 rows each) per wave

__device__ __forceinline__ float fast_sigmoid(float x) {
    // v_exp_f32 + v_rcp_f32; ~1 ulp rcp is plenty inside a sigmoid
    return __builtin_amdgcn_rcpf(1.0f + __expf(-x));
}
__device__ __forceinline__ float fast_tanh(float x) {
    // tanh(x) = 1 - 2/(exp(2x)+1); saturates correctly at +-1
    return 1.0f - 2.0f * __builtin_amdgcn_rcpf(__expf(2.0f * x) + 1.0f);
}

__global__ __launch_bounds__(32) void lstm_wmma_kernel(
    const float* __restrict__ x,     // [B,T,5]
    const float* __restrict__ h0,    // [1,B,10]
    const float* __restrict__ c0,    // [1,B,10]
    const float* __restrict__ W_ih,  // [40,5]
    const float* __restrict__ W_hh,  // [40,10]
    const float* __restrict__ b_ih,  // [40]
    const float* __restrict__ b_hh,  // [40]
    float* __restrict__ out)         // [B,T,10]
{
    // Wave-private h tiles in row-major [16 rows][16 cols]; DS ops are
    // in-order within a wave, so write(t) -> read(t+1) needs no barrier.
    __shared__ float h_lds[NT][16][16];

    const int lane = threadIdx.x;   // 0..31
    const int r    = lane & 15;     // batch row within tile (A layout: M)
    const int n    = lane & 15;     // output column (B/C/D layout: N)
    const int hi   = lane >> 4;     // lane half selects K sub-pair

    // ---- One-time: B-operands (transposed, gate-padded weights) + biases ----
    // Combined K: k<10 -> W_hh[.,k]; 10<=k<15 -> W_ih[.,k-10]; k=15 -> 0.
    // B operand (4x16, 2 VGPRs): lanes<16 hold rows k0,k0+1; lanes>=16 k0+2,k0+3.
    v2f   Bop[4][4];
    float bias[4];
#pragma unroll
    for (int g = 0; g < 4; ++g) {
#pragma unroll
        for (int c = 0; c < 4; ++c) {
            const int k0 = 4 * c + 2 * hi;   // even: never straddles k=10
            float w0 = 0.0f, w1 = 0.0f;
            if (n < H_SZ) {
                const int row = g * H_SZ + n;
                if (k0 < H_SZ) {
                    w0 = W_hh[row * H_SZ + k0];
                    w1 = W_hh[row * H_SZ + k0 + 1];
                } else if (k0 < H_SZ + I_SZ) {
                    w0 = W_ih[row * I_SZ + (k0 - H_SZ)];
                    w1 = (k0 + 1 < H_SZ + I_SZ) ? W_ih[row * I_SZ + (k0 + 1 - H_SZ)] : 0.0f;
                }
            }
            v2f b; b.x = w0; b.y = w1;
            Bop[g][c] = b;
        }
        bias[g] = (n < H_SZ) ? (b_ih[g * H_SZ + n] + b_hh[g * H_SZ + n]) : 0.0f;
    }
    v8f biasv[4];
#pragma unroll
    for (int g = 0; g < 4; ++g)
#pragma unroll
        for (int e = 0; e < 8; ++e) biasv[g][e] = bias[g];

    // ---- Per-tile init: state from h0/c0 (C/D layout), x row pointers ----
    int b0[NT];
    const float* xrow[NT];
    v8f c_st[NT];
#pragma unroll
    for (int u = 0; u < NT; ++u) {
        b0[u]   = blockIdx.x * (16 * NT) + u * 16;
        xrow[u] = x + (size_t)(b0[u] + r) * T_SZ * I_SZ;
#pragma unroll
        for (int e = 0; e < 8; ++e) {
            const int m = e + 8 * hi;                     // batch row in tile
            c_st[u][e] = (n < H_SZ) ? c0[(size_t)(b0[u] + m) * H_SZ + n] : 0.0f;
            h_lds[u][m][n] = (n < H_SZ) ? h0[(size_t)(b0[u] + m) * H_SZ + n] : 0.0f;
        }
    }

    // ---- Sequential scan over T ----
    for (int t = 0; t < T_SZ; ++t) {
#pragma unroll
        for (int u = 0; u < NT; ++u) {
            const float* xt = xrow[u] + t * I_SZ;
            // Per-lane x elements for A chunks 2/3 (see mapping below)
            const float xa = xt[hi ? 0 : 2];
            const float xb = xt[hi ? 1 : 3];
            const float x4 = xt[4];
            if (t + 8 < T_SZ) __builtin_prefetch(xt + 8 * I_SZ, 0, 3);

            // A = [h(10)|x(5)|0], 16x16, as 4 chunks of 16x4.
            // chunk c: lanes<16 hold k=4c,4c+1 of row r; lanes>=16 hold 4c+2,4c+3.
            const float* hrow = &h_lds[u][r][0];
            const v2f hp0 = *(const v2f*)(hrow + (hi ? 2 : 0));   // h0,1 / h2,3
            const v2f hp1 = *(const v2f*)(hrow + (hi ? 6 : 4));   // h4,5 / h6,7
            const v2f hp2 = *(const v2f*)(hrow + 8);              // h8,9 (lo lanes)
            v2f a1 = hp1, a2, a3;
            const v2f a0 = hp0;
            a2.x = hi ? xa : hp2.x;   // lo: h8      hi: x0
            a2.y = hi ? xb : hp2.y;   // lo: h9      hi: x1
            a3.x = hi ? x4 : xa;      // lo: x2      hi: x4
            a3.y = hi ? 0.0f : xb;    // lo: x3      hi: 0

            // gates[g] = ((bias + A0*B0) + A1*B1 + A2*B2 + A3*B3)
            // First WMMA of each chain reads the bias splat as C directly.
            v8f acc0 = __builtin_amdgcn_wmma_f32_16x16x4_f32(false, a0, false, Bop[0][0],
                                                             (short)0, biasv[0], false, false);
            v8f acc1 = __builtin_amdgcn_wmma_f32_16x16x4_f32(false, a0, false, Bop[1][0],
                                                             (short)0, biasv[1], false, false);
            v8f acc2 = __builtin_amdgcn_wmma_f32_16x16x4_f32(false, a0, false, Bop[2][0],
                                                             (short)0, biasv[2], false, false);
            v8f acc3 = __builtin_amdgcn_wmma_f32_16x16x4_f32(false, a0, false, Bop[3][0],
                                                             (short)0, biasv[3], false, false);
#pragma unroll
            for (int c = 1; c < 4; ++c) {
                const v2f a = (c == 1) ? a1 : (c == 2) ? a2 : a3;
                acc0 = __builtin_amdgcn_wmma_f32_16x16x4_f32(false, a, false, Bop[0][c],
                                                             (short)0, acc0, false, false);
                acc1 = __builtin_amdgcn_wmma_f32_16x16x4_f32(false, a, false, Bop[1][c],
                                                             (short)0, acc1, false, false);
                acc2 = __builtin_amdgcn_wmma_f32_16x16x4_f32(false, a, false, Bop[2][c],
                                                             (short)0, acc2, false, false);
                acc3 = __builtin_amdgcn_wmma_f32_16x16x4_f32(false, a, false, Bop[3][c],
                                                             (short)0, acc3, false, false);
            }

            // Elementwise gate math — identical C/D layout for all four tiles.
            v8f hnew;
#pragma unroll
            for (int e = 0; e < 8; ++e) {
                const float ig = fast_sigmoid(acc0[e]);
                const float fg = fast_sigmoid(acc1[e]);
                const float gg = fast_tanh(acc2[e]);
                const float og = fast_sigmoid(acc3[e]);
                const float cc = fg * c_st[u][e] + ig * gg;
                c_st[u][e] = cc;
                hnew[e] = og * fast_tanh(cc);
            }

            // Emit h: D layout -> LDS (next-step A source) + global out (NT hint).
#pragma unroll
            for (int v = 0; v < 8; ++v) {
                const int m = v + 8 * hi;
                h_lds[u][m][n] = hnew[v];
                if (n < H_SZ) {
                    float* op = out + ((size_t)(b0[u] + m) * T_SZ + t) * H_SZ + n;
                    __builtin_nontemporal_store(hnew[v], op);
                }
            }
        }
    }
}

extern "C" void kernel_launch(void* const* d_in, const int* in_sizes, int n_in,
                              void* d_out, int out_size, void* d_ws, size_t ws_size,
                              hipStream_t stream) {
    (void)in_sizes; (void)n_in; (void)d_ws; (void)ws_size; (void)out_size;
    const float* x    = (const float*)d_in[0];
    const float* h0   = (const float*)d_in[1];
    const float* c0   = (const float*)d_in[2];
    const float* W_ih = (const float*)d_in[3];
    const float* W_hh = (const float*)d_in[4];
    const float* b_ih = (const float*)d_in[5];
    const float* b_hh = (const float*)d_in[6];
    float* out = (float*)d_out;

    dim3 grid(B_SZ / (16 * NT));   // 64 single-wave workgroups
    dim3 block(32);
    hipLaunchKernelGGL(lstm_wmma_kernel, grid, block, 0, stream,
                       x, h0, c0, W_ih, W_hh, b_ih, b_hh, out);
}